// GINGraphLev_62130996904044
// MI455X (gfx1250) — compile-verified
//
#include <hip/hip_runtime.h>
#include <hip/hip_bf16.h>

typedef float v2f __attribute__((ext_vector_type(2)));
typedef float v8f __attribute__((ext_vector_type(8)));

// ---------------------------------------------------------------------------
// buf[i] = (1 + eps[0]) * x[i]
// ---------------------------------------------------------------------------
__global__ void k_scale_init(float* __restrict__ out, const float* __restrict__ x,
                             const float* __restrict__ eps, long long n) {
    long long i = (long long)blockIdx.x * blockDim.x + threadIdx.x;
    if (i < n) out[i] = (1.0f + eps[0]) * x[i];
}

// ---------------------------------------------------------------------------
// agg[dst[e]*F + f..f+3] += feat[src[e]*F + f..f+3]
// one thread per (edge, 4-feature group): 1x b128 load + 4x atomic_add_f32
// ---------------------------------------------------------------------------
__global__ void k_scatter_add4(const float* __restrict__ feat, const int* __restrict__ srcI,
                               const int* __restrict__ dstI, float* __restrict__ agg,
                               int E, int Fq /* = F/4 */) {
    long long i = (long long)blockIdx.x * blockDim.x + threadIdx.x;
    long long total = (long long)E * Fq;
    if (i >= total) return;
    int e = (int)(i / Fq);
    int q = (int)(i - (long long)e * Fq);
    int s = srcI[e];
    int d = dstI[e];
    const float4 v = *(const float4*)(feat + ((size_t)s * Fq + q) * 4);
    float* p = agg + ((size_t)d * Fq + q) * 4;
    atomicAdd(p + 0, v.x);
    atomicAdd(p + 1, v.y);
    atomicAdd(p + 2, v.z);
    atomicAdd(p + 3, v.w);
}

__global__ void k_zero(float* __restrict__ p, long long n) {
    long long i = (long long)blockIdx.x * blockDim.x + threadIdx.x;
    if (i < n) p[i] = 0.0f;
}

// ---------------------------------------------------------------------------
// C[r,c] = act( sum_k A[r,k] * W[k,c] + bias[c] )   fp32 WMMA 16x16x4
// Block = 128 thr (4 waves) -> 64 rows x 64 cols. W panel staged in LDS,
// K-pair interleaved [k/2][col][2] so a B fragment is one ds_load_b64.
// A frag : lane l holds A[row0 + (l&15)][k + 2*(l>>4) + v]   (v2f load)
// B frag : lane l holds W[k + 2*(l>>4) + v][col0 + (l&15)]
// D frag : lane l, elem v -> row0 + v + 8*(l>>4), col0 + (l&15)
// NOTE: nRows must be a multiple of 16 (N = 50000 = 3125*16): every wave that
// passes the uniform row0 guard owns 16 fully in-range rows, so the epilogue
// stores are unconditional.
// ---------------------------------------------------------------------------
template <int K, int HOUT, bool RELU>
__global__ void k_gemm_wmma(const float* __restrict__ A, const float* __restrict__ W,
                            const float* __restrict__ bias, float* __restrict__ C,
                            int nRows) {
    __shared__ float ldsW[(K / 2) * 128];   // [k/2][col 0..63][2]

    const int tid  = threadIdx.x;
    const int col0 = blockIdx.y * 64;

    // Cooperative fill: b128 global loads, pair-interleaved LDS stores.
    for (int idx = tid; idx < K * 16; idx += 128) {
        int kk = idx >> 4;          // 0..K-1
        int cq = idx & 15;          // quad of columns
        const float4 w = *(const float4*)(W + (size_t)kk * HOUT + col0 + cq * 4);
        float* dst = ldsW + (kk >> 1) * 128 + (kk & 1);
        dst[(cq * 4 + 0) * 2] = w.x;
        dst[(cq * 4 + 1) * 2] = w.y;
        dst[(cq * 4 + 2) * 2] = w.z;
        dst[(cq * 4 + 3) * 2] = w.w;
    }
    __syncthreads();

    const int wave = tid >> 5;
    const int lane = tid & 31;
    const int row0 = (blockIdx.x * 4 + wave) * 16;
    if (row0 >= nRows) return;                // wave-uniform: EXEC stays all-1s
    const int h = lane >> 4;                  // lane half (0/1)
    const int m = lane & 15;

    const float* Arow = A + (size_t)(row0 + m) * K + 2 * h;

    v8f acc[4] = {};

    for (int k = 0; k < K; k += 4) {
        v2f a = *(const v2f*)(Arow + k);                 // 8B aligned
        const float* Bp = ldsW + ((k >> 1) + h) * 128 + m * 2;
#pragma unroll
        for (int t = 0; t < 4; ++t) {
            v2f b = *(const v2f*)(Bp + t * 32);          // ds_load_b64
            acc[t] = __builtin_amdgcn_wmma_f32_16x16x4_f32(
                false, a, false, b, (short)0, acc[t], false, false);
        }
    }

#pragma unroll
    for (int t = 0; t < 4; ++t) {
        int col = col0 + t * 16 + m;
        float bc = bias[col];
        float* Crow = C + (size_t)(row0 + 8 * h) * HOUT + col;
#pragma unroll
        for (int v = 0; v < 8; ++v) {
            float val = acc[t][v] + bc;
            if (RELU) val = fmaxf(val, 0.0f);
            Crow[(size_t)v * HOUT] = val;
        }
    }
}

// ---------------------------------------------------------------------------
// psum[batch[n]*F + f..f+3] += h2[n*F + f..f+3];  cnt[batch[n]] += 1 per node
// ---------------------------------------------------------------------------
__global__ void k_pool4(const float* __restrict__ h2, const int* __restrict__ batch,
                        float* __restrict__ psum, float* __restrict__ cnt,
                        int Nn, int Fq /* = F/4 */) {
    long long i = (long long)blockIdx.x * blockDim.x + threadIdx.x;
    long long total = (long long)Nn * Fq;
    if (i >= total) return;
    int n = (int)(i / Fq);
    int q = (int)(i - (long long)n * Fq);
    int g = batch[n];
    const float4 v = *(const float4*)(h2 + ((size_t)n * Fq + q) * 4);
    float* p = psum + ((size_t)g * Fq + q) * 4;
    atomicAdd(p + 0, v.x);
    atomicAdd(p + 1, v.y);
    atomicAdd(p + 2, v.z);
    atomicAdd(p + 3, v.w);
    if (q == 0) atomicAdd(&cnt[g], 1.0f);
}

// ---------------------------------------------------------------------------
// out[g,:] = log_softmax( (psum[g,:]/max(cnt[g],1)) @ Wlin + blin )
// ---------------------------------------------------------------------------
__global__ void k_final(const float* __restrict__ psum, const float* __restrict__ cnt,
                        const float* __restrict__ Wlin, const float* __restrict__ blin,
                        float* __restrict__ out, int Gn, int F, int O) {
    int g = blockIdx.x * blockDim.x + threadIdx.x;
    if (g >= Gn) return;
    float invc = 1.0f / fmaxf(cnt[g], 1.0f);
    float z[16];
    for (int o = 0; o < O; ++o) z[o] = blin[o];
    for (int f = 0; f < F; ++f) {
        float p = psum[(long long)g * F + f] * invc;
        for (int o = 0; o < O; ++o) z[o] += p * Wlin[(long long)f * O + o];
    }
    float mx = z[0];
    for (int o = 1; o < O; ++o) mx = fmaxf(mx, z[o]);
    float s = 0.0f;
    for (int o = 0; o < O; ++o) s += expf(z[o] - mx);
    float lse = mx + logf(s);
    for (int o = 0; o < O; ++o) out[(long long)g * O + o] = z[o] - lse;
}

// ---------------------------------------------------------------------------
extern "C" void kernel_launch(void* const* d_in, const int* in_sizes, int n_in,
                              void* d_out, int out_size, void* d_ws, size_t ws_size,
                              hipStream_t stream) {
    const int Nn = 50000, Ee = 800000, INf = 128, Hf = 256, OUTf = 10, Gg = 512;

    const float* x    = (const float*)d_in[0];
    const int*   ei   = (const int*)d_in[1];
    const int*   bat  = (const int*)d_in[2];
    const float* eps1 = (const float*)d_in[3];
    const float* W1a  = (const float*)d_in[4];
    const float* b1a  = (const float*)d_in[5];
    const float* W1b  = (const float*)d_in[6];
    const float* b1b  = (const float*)d_in[7];
    const float* eps2 = (const float*)d_in[8];
    const float* W2a  = (const float*)d_in[9];
    const float* b2a  = (const float*)d_in[10];
    const float* W2b  = (const float*)d_in[11];
    const float* b2b  = (const float*)d_in[12];
    const float* Wlin = (const float*)d_in[13];
    const float* blin = (const float*)d_in[14];
    float* out = (float*)d_out;

    const int* srcI = ei;        // edge_index[0]
    const int* dstI = ei + Ee;   // edge_index[1]

    // Workspace layout (floats): agg1[N*IN] | T[N*H] | h[N*H] | h2buf[N*H]
    float* wsf   = (float*)d_ws;
    float* agg1  = wsf;
    float* T     = agg1 + (size_t)Nn * INf;
    float* hbuf  = T    + (size_t)Nn * Hf;
    float* h2buf = hbuf + (size_t)Nn * Hf;
    float* psum  = agg1;                     // reuse: free after GEMM1
    float* cnt   = psum + (size_t)Gg * Hf;

    dim3 blk(128);
    dim3 gGemm((Nn / 16 + 3) / 4, Hf / 64);  // 782 x 4 blocks, 4 waves/block

    // --- GIN layer 1 ---
    long long n1 = (long long)Nn * INf;
    k_scale_init<<<(int)((n1 + 255) / 256), 256, 0, stream>>>(agg1, x, eps1, n1);
    long long eq1 = (long long)Ee * (INf / 4);
    k_scatter_add4<<<(int)((eq1 + 255) / 256), 256, 0, stream>>>(x, srcI, dstI, agg1, Ee, INf / 4);
    k_gemm_wmma<128, 256, true><<<gGemm, blk, 0, stream>>>(agg1, W1a, b1a, T, Nn);
    k_gemm_wmma<256, 256, true><<<gGemm, blk, 0, stream>>>(T, W1b, b1b, hbuf, Nn); // outer relu fused

    // --- GIN layer 2 ---
    long long n2 = (long long)Nn * Hf;
    k_scale_init<<<(int)((n2 + 255) / 256), 256, 0, stream>>>(h2buf, hbuf, eps2, n2);
    long long eq2 = (long long)Ee * (Hf / 4);
    k_scatter_add4<<<(int)((eq2 + 255) / 256), 256, 0, stream>>>(hbuf, srcI, dstI, h2buf, Ee, Hf / 4);
    k_gemm_wmma<256, 256, true><<<gGemm, blk, 0, stream>>>(h2buf, W2a, b2a, T, Nn);
    k_gemm_wmma<256, 256, false><<<gGemm, blk, 0, stream>>>(T, W2b, b2b, h2buf, Nn);

    // --- mean pool + classifier + log_softmax ---
    long long np = (long long)Gg * Hf + Gg;
    k_zero<<<(int)((np + 255) / 256), 256, 0, stream>>>(psum, np);
    long long nq = (long long)Nn * (Hf / 4);
    k_pool4<<<(int)((nq + 255) / 256), 256, 0, stream>>>(h2buf, bat, psum, cnt, Nn, Hf / 4);
    k_final<<<2, 256, 0, stream>>>(psum, cnt, Wlin, blin, out, Gg, Hf, OUTf);
}